// DLN_2190433321000
// MI455X (gfx1250) — compile-verified
//
#include <hip/hip_runtime.h>
#include <math.h>

// Problem constants (match the reference).
#define T_LEN 8192
#define D_CH  2048
#define CHUNK 128              // L: rows per chunk
#define NCHUNK (T_LEN / CHUNK) // C = 64
#define TPB   256              // threads per block
#define PAIRS (D_CH / 2)       // 1024 channel-pairs (2 channels per thread)
#define BLK_PER_CHUNK (D_CH / (TPB * 2)) // 4

// Native vector types for builtins.
typedef int   v2i __attribute__((ext_vector_type(2)));
typedef float v4f __attribute__((ext_vector_type(4)));

// gfx1250 async global->LDS staging (ASYNCcnt-tracked), if the toolchain
// exposes the builtin. Fallback: direct loads + global_prefetch_b8.
#if defined(__gfx1250__) && __has_builtin(__builtin_amdgcn_global_load_async_to_lds_b64)
#define HAVE_ASYNC_LDS 1
#else
#define HAVE_ASYNC_LDS 0
#endif

#if HAVE_ASYNC_LDS
typedef __attribute__((address_space(1))) v2i* gptr_v2i;
typedef __attribute__((address_space(3))) v2i* lptr_v2i;

__device__ __forceinline__ void async_stage_b64(const float2* gsrc, float2* lds_dst) {
    __builtin_amdgcn_global_load_async_to_lds_b64(
        (gptr_v2i)(v2i*)gsrc, (lptr_v2i)(v2i*)lds_dst, 0, 0);
}

#if __has_builtin(__builtin_amdgcn_s_wait_asynccnt)
#define WAIT_ASYNC_LE1() __builtin_amdgcn_s_wait_asynccnt(1)
#else
#define WAIT_ASYNC_LE1() asm volatile("s_wait_asynccnt 0x1" ::: "memory")
#endif
#endif // HAVE_ASYNC_LDS

__device__ __forceinline__ void nt_store4(float a, float b, float c, float d, float4* p) {
    v4f v = (v4f){a, b, c, d};
    __builtin_nontemporal_store(v, (v4f*)p);
}

// Per-channel z = exp(-exp(size)) * (cos(theta) + i sin(theta))
__device__ __forceinline__ void make_z(float sz, float th, float& zr, float& zi) {
    float r = __expf(-__expf(sz));
    float s, c;
    __sincosf(th, &s, &c);
    zr = r * c;
    zi = r * s;
}

// z^L closed form: |z|^L = exp(-L*exp(size)), arg = L*theta
__device__ __forceinline__ void make_zL(float sz, float th, float& zr, float& zi) {
    float r = __expf(-(float)CHUNK * __expf(sz));
    float s, c;
    __sincosf((float)CHUNK * th, &s, &c);
    zr = r * c;
    zi = r * s;
}

// ---------------------------------------------------------------------------
// Kernel 1: per (chunk, channel-pair) compute the chunk-local end state.
// x loads (via async-to-LDS) populate L2 so kernel 3's second pass hits L2.
// ---------------------------------------------------------------------------
__global__ void __launch_bounds__(TPB)
scan_chunk_ends(const float* __restrict__ x,
                const float* __restrict__ size_,
                const float* __restrict__ theta,
                float4* __restrict__ ends) {
    const int pair = blockIdx.x * TPB + threadIdx.x;   // 0..PAIRS-1
    const int c    = blockIdx.y;                       // chunk index
    const int d0   = pair * 2;

    float zr0, zi0, zr1, zi1;
    make_z(size_[d0],     theta[d0],     zr0, zi0);
    make_z(size_[d0 + 1], theta[d0 + 1], zr1, zi1);

    const float2* __restrict__ xp =
        (const float2*)x + (size_t)c * CHUNK * PAIRS + pair;

    float vr0 = 0.f, vi0 = 0.f, vr1 = 0.f, vi1 = 0.f;

#if HAVE_ASYNC_LDS
    __shared__ float2 stage[2][TPB];
    async_stage_b64(xp, &stage[0][threadIdx.x]);
    for (int t = 0; t < CHUNK; ++t) {
        if (t + 1 < CHUNK)
            async_stage_b64(xp + (size_t)(t + 1) * PAIRS, &stage[(t + 1) & 1][threadIdx.x]);
        WAIT_ASYNC_LE1();                      // row t staged; t+1 in flight
        float2 xv = stage[t & 1][threadIdx.x];
        float nr0 = fmaf(zr0, vr0, fmaf(-zi0, vi0, xv.x));
        float ni0 = fmaf(zr0, vi0, zi0 * vr0);
        float nr1 = fmaf(zr1, vr1, fmaf(-zi1, vi1, xv.y));
        float ni1 = fmaf(zr1, vi1, zi1 * vr1);
        vr0 = nr0; vi0 = ni0; vr1 = nr1; vi1 = ni1;
    }
#else
#pragma unroll 8
    for (int t = 0; t < CHUNK; ++t) {
        __builtin_prefetch(xp + (size_t)(t + 24) * PAIRS, 0, 0);
        float2 xv = xp[(size_t)t * PAIRS];
        float nr0 = fmaf(zr0, vr0, fmaf(-zi0, vi0, xv.x));
        float ni0 = fmaf(zr0, vi0, zi0 * vr0);
        float nr1 = fmaf(zr1, vr1, fmaf(-zi1, vi1, xv.y));
        float ni1 = fmaf(zr1, vi1, zi1 * vr1);
        vr0 = nr0; vi0 = ni0; vr1 = nr1; vi1 = ni1;
    }
#endif

    ends[(size_t)c * PAIRS + pair] = make_float4(vr0, vi0, vr1, vi1);
}

// ---------------------------------------------------------------------------
// Kernel 2: per channel-pair, exclusive scan over the C chunk aggregates:
//   carry[0] = 0;  carry[c] = z^L * carry[c-1] + ends[c-1]
// ---------------------------------------------------------------------------
__global__ void __launch_bounds__(TPB)
scan_carries(const float* __restrict__ size_,
             const float* __restrict__ theta,
             const float4* __restrict__ ends,
             float4* __restrict__ carries) {
    const int pair = blockIdx.x * TPB + threadIdx.x;   // 0..PAIRS-1
    if (pair >= PAIRS) return;
    const int d0 = pair * 2;

    float zr0, zi0, zr1, zi1;
    make_zL(size_[d0],     theta[d0],     zr0, zi0);
    make_zL(size_[d0 + 1], theta[d0 + 1], zr1, zi1);

    float sr0 = 0.f, si0 = 0.f, sr1 = 0.f, si1 = 0.f;
    for (int c = 0; c < NCHUNK; ++c) {
        carries[(size_t)c * PAIRS + pair] = make_float4(sr0, si0, sr1, si1);
        float4 e = ends[(size_t)c * PAIRS + pair];
        float nr0 = fmaf(zr0, sr0, fmaf(-zi0, si0, e.x));
        float ni0 = fmaf(zr0, si0, fmaf( zi0, sr0, e.y));
        float nr1 = fmaf(zr1, sr1, fmaf(-zi1, si1, e.z));
        float ni1 = fmaf(zr1, si1, fmaf( zi1, sr1, e.w));
        sr0 = nr0; si0 = ni0; sr1 = nr1; si1 = ni1;
    }
}

// ---------------------------------------------------------------------------
// Kernel 3: re-scan each chunk seeded with its carry; x reads hit L2 (loaded
// by kernel 1). Output (128MB) is written with non-temporal stores so the
// write stream does not evict x from L2. Each channel-pair's two complexes
// per row form one 16B-aligned float4.
// ---------------------------------------------------------------------------
__global__ void __launch_bounds__(TPB)
scan_write(const float* __restrict__ x,
           const float* __restrict__ size_,
           const float* __restrict__ theta,
           const float4* __restrict__ carries,
           float4* __restrict__ out4) {
    const int pair = blockIdx.x * TPB + threadIdx.x;   // 0..PAIRS-1
    const int c    = blockIdx.y;
    const int d0   = pair * 2;

    float zr0, zi0, zr1, zi1;
    make_z(size_[d0],     theta[d0],     zr0, zi0);
    make_z(size_[d0 + 1], theta[d0 + 1], zr1, zi1);

    float4 s = carries[(size_t)c * PAIRS + pair];
    float vr0 = s.x, vi0 = s.y, vr1 = s.z, vi1 = s.w;

    const float2* __restrict__ xp =
        (const float2*)x + (size_t)c * CHUNK * PAIRS + pair;
    float4* __restrict__ op = out4 + (size_t)c * CHUNK * PAIRS + pair;

#if HAVE_ASYNC_LDS
    __shared__ float2 stage[2][TPB];
    async_stage_b64(xp, &stage[0][threadIdx.x]);
    for (int t = 0; t < CHUNK; ++t) {
        if (t + 1 < CHUNK)
            async_stage_b64(xp + (size_t)(t + 1) * PAIRS, &stage[(t + 1) & 1][threadIdx.x]);
        WAIT_ASYNC_LE1();
        float2 xv = stage[t & 1][threadIdx.x];
        float nr0 = fmaf(zr0, vr0, fmaf(-zi0, vi0, xv.x));
        float ni0 = fmaf(zr0, vi0, zi0 * vr0);
        float nr1 = fmaf(zr1, vr1, fmaf(-zi1, vi1, xv.y));
        float ni1 = fmaf(zr1, vi1, zi1 * vr1);
        vr0 = nr0; vi0 = ni0; vr1 = nr1; vi1 = ni1;
        nt_store4(vr0, vi0, vr1, vi1, op + (size_t)t * PAIRS);
    }
#else
#pragma unroll 8
    for (int t = 0; t < CHUNK; ++t) {
        __builtin_prefetch(xp + (size_t)(t + 24) * PAIRS, 0, 0);
        float2 xv = xp[(size_t)t * PAIRS];
        float nr0 = fmaf(zr0, vr0, fmaf(-zi0, vi0, xv.x));
        float ni0 = fmaf(zr0, vi0, zi0 * vr0);
        float nr1 = fmaf(zr1, vr1, fmaf(-zi1, vi1, xv.y));
        float ni1 = fmaf(zr1, vi1, zi1 * vr1);
        vr0 = nr0; vi0 = ni0; vr1 = nr1; vi1 = ni1;
        nt_store4(vr0, vi0, vr1, vi1, op + (size_t)t * PAIRS);
    }
#endif
}

extern "C" void kernel_launch(void* const* d_in, const int* in_sizes, int n_in,
                              void* d_out, int out_size, void* d_ws, size_t ws_size,
                              hipStream_t stream) {
    const float* x     = (const float*)d_in[0];   // [T, D] f32
    const float* size_ = (const float*)d_in[1];   // [D]    f32
    const float* theta = (const float*)d_in[2];   // [D]    f32

    // Workspace: ends[C][PAIRS] float4, then carries[C][PAIRS] float4 (2 MB).
    float4* ends    = (float4*)d_ws;
    float4* carries = ends + (size_t)NCHUNK * PAIRS;

    float4* out4 = (float4*)d_out;                // complex64 [T, D] interleaved

    dim3 grid1(BLK_PER_CHUNK, NCHUNK);
    scan_chunk_ends<<<grid1, TPB, 0, stream>>>(x, size_, theta, ends);

    scan_carries<<<(PAIRS + TPB - 1) / TPB, TPB, 0, stream>>>(size_, theta, ends, carries);

    dim3 grid3(BLK_PER_CHUNK, NCHUNK);
    scan_write<<<grid3, TPB, 0, stream>>>(x, size_, theta, carries, out4);
}